// Net_1039382085697
// MI455X (gfx1250) — compile-verified
//
#include <hip/hip_runtime.h>
#include <hip/hip_bf16.h>

typedef __attribute__((ext_vector_type(2))) float v2f;
typedef __attribute__((ext_vector_type(8))) float v8f;

#define NN     100000
#define EE     2000000LL
#define DIM    64
#define F_IN   128
#define RR     64
#define RF     64
#define ERE    2048
#define NCLS   16

// ---------------- utility kernels ----------------

__global__ void k_zero(float* __restrict__ p, long long n) {
  long long i = (long long)blockIdx.x * blockDim.x + threadIdx.x;
  if (i < n) p[i] = 0.0f;
}

__global__ void k_deg(const int* __restrict__ ei, float* __restrict__ deg, long long E) {
  long long e = (long long)blockIdx.x * blockDim.x + threadIdx.x;
  if (e < E) atomicAdd(&deg[ei[E + e]], 1.0f);   // row 1 of edge_index = dst
}

__global__ void k_recip(float* __restrict__ deg, int n) {
  int i = blockIdx.x * blockDim.x + threadIdx.x;
  if (i < n) deg[i] = 1.0f / fmaxf(deg[i], 1.0f);
}

// ---------------- relation conv (x2) + W_per_rel = rel_h @ W_nn + b_nn ----------------
// Single workgroup; tiny problem (R=64, RF=64, ER=2048). Final GEMM uses fp32 WMMA.

__global__ void k_rel(const float* __restrict__ rel_x, const int* __restrict__ rel_ei,
                      const float* __restrict__ rel_ea, const float* __restrict__ W_nn,
                      const float* __restrict__ b_nn, float* __restrict__ W_per_rel) {
  __shared__ float h[RR * RF];    // 16 KB
  __shared__ float s[RR * RF];    // 16 KB
  __shared__ float cnt[RR];
  const int tid = threadIdx.x;    // 256 threads
  for (int i = tid; i < RR * RF; i += 256) h[i] = rel_x[i];
  const int* srcr = rel_ei;
  const int* dstr = rel_ei + ERE;

  for (int it = 0; it < 2; ++it) {
    for (int i = tid; i < RR * RF; i += 256) s[i] = 0.0f;
    if (tid < RR) cnt[tid] = 0.0f;
    __syncthreads();
    // 4 threads per edge, 16 features each
    for (int e4 = tid; e4 < ERE * 4; e4 += 256) {
      int e = e4 >> 2, fo = (e4 & 3) * 16;
      int sN = srcr[e], dN = dstr[e];
      float a = rel_ea[e];
      #pragma unroll
      for (int f = 0; f < 16; ++f)
        atomicAdd(&s[dN * RF + fo + f], h[sN * RF + fo + f] * a);
      if ((e4 & 3) == 0) atomicAdd(&cnt[dN], 1.0f);
    }
    __syncthreads();
    for (int i = tid; i < RR * RF; i += 256) {
      int r = i >> 6;  // /RF
      float v = h[i] + s[i] / fmaxf(cnt[r], 1.0f);
      h[i] = fmaxf(v, 0.0f);   // relu
    }
    __syncthreads();
  }

  // W_per_rel[64,64] = h @ W_nn + b_nn via V_WMMA_F32_16X16X4_F32.
  // 16 output tiles of 16x16; 8 waves -> 2 tiles per wave (uniform control flow, EXEC all-1s).
  const int wave = tid >> 5, lane = tid & 31;
  const int koff = (lane < 16) ? 0 : 2;
  const int l15 = lane & 15;
  for (int t = wave; t < 16; t += 8) {
    const int tm = (t >> 2) * 16, tn = (t & 3) * 16;
    v8f acc = {};
    for (int k = 0; k < RF; k += 4) {
      v2f a, b;
      a.x = h[(tm + l15) * RF + k + koff];
      a.y = h[(tm + l15) * RF + k + koff + 1];
      b.x = W_nn[(k + koff) * DIM + tn + l15];
      b.y = W_nn[(k + koff + 1) * DIM + tn + l15];
      acc = __builtin_amdgcn_wmma_f32_16x16x4_f32(false, a, false, b, (short)0, acc, false, false);
    }
    const int rbase = tm + ((lane < 16) ? 0 : 8);
    const float bias = b_nn[tn + l15];
    #pragma unroll
    for (int r = 0; r < 8; ++r)
      W_per_rel[(rbase + r) * DIM + tn + l15] = acc[r] + bias;
  }
}

// ---------------- x = x_in @ fc1  ([N,128]x[128,64]) fp32 WMMA ----------------
// One 16x64 output tile per wave (4 v8f accumulators); fc1 staged in LDS (32 KB).

__global__ void k_fc1(const float* __restrict__ xin, const float* __restrict__ fc1,
                      float* __restrict__ x, int nrows) {
  __shared__ float Bs[F_IN * DIM];
  const int tid = threadIdx.x;   // 256
  for (int i = tid; i < F_IN * DIM; i += 256) Bs[i] = fc1[i];
  __syncthreads();

  const int wave = tid >> 5, lane = tid & 31;
  const int tileM = blockIdx.x * 8 + wave;
  if (tileM * 16 >= nrows) return;      // uniform per wave -> EXEC stays all-1s
  const int l15 = lane & 15;
  const int koff = (lane < 16) ? 0 : 2;
  const float* arow = xin + (size_t)(tileM * 16 + l15) * F_IN;

  v8f acc[4] = {};
  for (int k = 0; k < F_IN; k += 4) {
    v2f a;
    a.x = arow[k + koff];
    a.y = arow[k + koff + 1];
    #pragma unroll
    for (int nt = 0; nt < 4; ++nt) {
      v2f b;
      b.x = Bs[(k + koff) * DIM + nt * 16 + l15];
      b.y = Bs[(k + koff + 1) * DIM + nt * 16 + l15];
      acc[nt] = __builtin_amdgcn_wmma_f32_16x16x4_f32(false, a, false, b, (short)0, acc[nt], false, false);
    }
  }
  const int rbase = tileM * 16 + ((lane < 16) ? 0 : 8);
  #pragma unroll
  for (int nt = 0; nt < 4; ++nt)
    #pragma unroll
    for (int r = 0; r < 8; ++r)
      x[(size_t)(rbase + r) * DIM + nt * 16 + l15] = acc[nt][r];
}

// ---------------- edge gather-multiply-scatter (the hot loop) ----------------
// 16 threads per edge, float4 per thread; x and W_per_rel are L2-resident.

__global__ void k_edge(const float* __restrict__ x, const int* __restrict__ ei,
                       const int* __restrict__ et, const float* __restrict__ Wpr,
                       float* __restrict__ agg, long long E) {
  long long gid = (long long)blockIdx.x * blockDim.x + threadIdx.x;
  if (gid >= E * 16) return;
  long long e = gid >> 4;
  int fo = (int)(gid & 15) * 4;
  int sN = ei[e];
  int dN = ei[E + e];
  int r  = et[e];
  const float4 xs = *(const float4*)(x   + (size_t)sN * DIM + fo);
  const float4 wv = *(const float4*)(Wpr + (size_t)r  * DIM + fo);
  float* out = agg + (size_t)dN * DIM + fo;
  atomicAdd(out + 0, xs.x * wv.x);
  atomicAdd(out + 1, xs.y * wv.y);
  atomicAdd(out + 2, xs.z * wv.z);
  atomicAdd(out + 3, xs.w * wv.w);
}

// ---------------- finalize: x = tanh(agg * recip + bias) ----------------

__global__ void k_fin(const float* __restrict__ agg, const float* __restrict__ recip,
                      const float* __restrict__ bias, float* __restrict__ x, long long n) {
  long long i = (long long)blockIdx.x * blockDim.x + threadIdx.x;
  if (i >= n) return;
  int node = (int)(i >> 6), f = (int)(i & 63);
  x[i] = tanhf(agg[i] * recip[node] + bias[f]);
}

// ---------------- out = x @ fc2_w + fc2_b  ([N,64]x[64,16]) fp32 WMMA ----------------

__global__ void k_fc2(const float* __restrict__ x, const float* __restrict__ w,
                      const float* __restrict__ b, float* __restrict__ out, int nrows) {
  const int tid = threadIdx.x, wave = tid >> 5, lane = tid & 31;
  const int tileM = blockIdx.x * 8 + wave;
  if (tileM * 16 >= nrows) return;
  const int l15 = lane & 15;
  const int koff = (lane < 16) ? 0 : 2;
  const float* arow = x + (size_t)(tileM * 16 + l15) * DIM;
  v8f acc = {};
  for (int k = 0; k < DIM; k += 4) {
    v2f a, bb;
    a.x  = arow[k + koff];
    a.y  = arow[k + koff + 1];
    bb.x = w[(k + koff) * NCLS + l15];
    bb.y = w[(k + koff + 1) * NCLS + l15];
    acc = __builtin_amdgcn_wmma_f32_16x16x4_f32(false, a, false, bb, (short)0, acc, false, false);
  }
  const int rbase = tileM * 16 + ((lane < 16) ? 0 : 8);
  const float bias = b[l15];
  #pragma unroll
  for (int r = 0; r < 8; ++r)
    out[(size_t)(rbase + r) * NCLS + l15] = acc[r] + bias;
}

// ---------------- host ----------------

extern "C" void kernel_launch(void* const* d_in, const int* in_sizes, int n_in,
                              void* d_out, int out_size, void* d_ws, size_t ws_size,
                              hipStream_t stream) {
  const float* x_in   = (const float*)d_in[0];
  const float* rel_x  = (const float*)d_in[1];
  const int*   ei     = (const int*)  d_in[2];
  const int*   et     = (const int*)  d_in[3];
  const int*   rel_ei = (const int*)  d_in[4];
  const float* rel_ea = (const float*)d_in[5];
  const float* fc1    = (const float*)d_in[6];
  const float* W_nn   = (const float*)d_in[7];
  const float* b_nn   = (const float*)d_in[8];
  const float* b_c1   = (const float*)d_in[9];
  const float* b_c2   = (const float*)d_in[10];
  const float* fc2_w  = (const float*)d_in[11];
  const float* fc2_b  = (const float*)d_in[12];
  float* out = (float*)d_out;

  const int N = in_sizes[0] / F_IN;     // 100000
  const long long E = in_sizes[3];      // 2000000

  // workspace layout
  float* x    = (float*)d_ws;           // N*64
  float* agg  = x   + (size_t)N * DIM;  // N*64
  float* deg  = agg + (size_t)N * DIM;  // N (recip degree)
  float* Wpr  = deg + N;                // 64*64

  const long long nfeat = (long long)N * DIM;
  const int TB = 256;
  auto blocks = [](long long n, int tb) { return (unsigned)((n + tb - 1) / tb); };

  // degree (shared by both convs): zero agg+deg contiguously, count, reciprocal
  k_zero <<<blocks(nfeat + N, TB), TB, 0, stream>>>(agg, nfeat + N);
  k_deg  <<<blocks(E, TB),         TB, 0, stream>>>(ei, deg, E);
  k_recip<<<blocks(N, TB),         TB, 0, stream>>>(deg, N);

  // relation convs + per-relation weight matrix (WMMA)
  k_rel<<<1, 256, 0, stream>>>(rel_x, rel_ei, rel_ea, W_nn, b_nn, Wpr);

  // input projection (WMMA)
  const unsigned gemm_blocks = (unsigned)((N / 16 + 7) / 8);
  k_fc1<<<gemm_blocks, 256, 0, stream>>>(x_in, fc1, x, N);

  // NNConv round 1
  k_edge<<<blocks(E * 16, TB), TB, 0, stream>>>(x, ei, et, Wpr, agg, E);
  k_fin <<<blocks(nfeat, TB),  TB, 0, stream>>>(agg, deg, b_c1, x, nfeat);

  // NNConv round 2
  k_zero<<<blocks(nfeat, TB),  TB, 0, stream>>>(agg, nfeat);
  k_edge<<<blocks(E * 16, TB), TB, 0, stream>>>(x, ei, et, Wpr, agg, E);
  k_fin <<<blocks(nfeat, TB),  TB, 0, stream>>>(agg, deg, b_c2, x, nfeat);

  // classifier head (WMMA)
  k_fc2<<<gemm_blocks, 256, 0, stream>>>(x, fc2_w, fc2_b, out, N);
}